// LearnableSingleMMF_1408749273719
// MI455X (gfx1250) — compile-verified
//
#include <hip/hip_runtime.h>

#define NN 2048

typedef __attribute__((ext_vector_type(2))) float v2f;
typedef __attribute__((ext_vector_type(8))) float v8f;

// ---------------------------------------------------------------------------
// init: Dw = A (working copy), right = I, active = 1
// ---------------------------------------------------------------------------
__global__ __launch_bounds__(256) void mmf_init_kernel(const float* __restrict__ A,
                                                       float* __restrict__ Dw,
                                                       float* __restrict__ right,
                                                       float* __restrict__ active) {
  size_t i = (size_t)blockIdx.x * blockDim.x + threadIdx.x;
  if (i >= (size_t)NN * NN) return;
  int row = (int)(i / NN);
  int col = (int)(i % NN);
  Dw[i] = A[i];
  right[i] = (row == col) ? 1.0f : 0.0f;
  if (i < (size_t)NN) active[i] = 1.0f;
}

__global__ __launch_bounds__(256) void mmf_wav_kernel(const int* __restrict__ wav,
                                                      float* __restrict__ active, int L) {
  int t = blockIdx.x * blockDim.x + threadIdx.x;
  if (t < L) active[wav[t]] = 0.0f;
}

// ---------------------------------------------------------------------------
// Sequential level sweep: 1 workgroup, 1024 threads, loops L levels.
//   phase1: A[idx,:]  = O @ A[idx,:]     (each thread owns 2 columns)
//   phase2: A[:,idx]  = A[:,idx] @ O^T   (each thread owns 2 rows)
//   phase3: right[idx,:] = O @ right[idx,:]
// ---------------------------------------------------------------------------
__global__ __launch_bounds__(1024) void mmf_levels_kernel(float* __restrict__ Aw,
                                                          float* __restrict__ right,
                                                          const float* __restrict__ O_all,
                                                          const int* __restrict__ sel,
                                                          int L) {
  __shared__ float Osh[16][16];
  __shared__ int idx[16];
  const int tid = threadIdx.x;

  for (int l = 0; l < L; ++l) {
    __syncthreads();  // prior level's phase2/3 writes visible; Osh reusable
    if (tid < 256) Osh[tid >> 4][tid & 15] = O_all[(size_t)l * 256 + tid];
    if (tid < 16) idx[tid] = sel[l * 16 + tid];
    __syncthreads();

    // ---- phase 1: rewrite 16 rows of Aw, this thread owns cols j, j+1 ----
    {
      const int j = tid * 2;
      float in0[16], in1[16];
#pragma unroll
      for (int k = 0; k < 16; ++k) {
        const float* p = &Aw[(size_t)idx[k] * NN + j];
        in0[k] = p[0];
        in1[k] = p[1];
      }
#pragma unroll
      for (int i = 0; i < 16; ++i) {
        float s0 = 0.0f, s1 = 0.0f;
#pragma unroll
        for (int k = 0; k < 16; ++k) {
          s0 = fmaf(Osh[i][k], in0[k], s0);
          s1 = fmaf(Osh[i][k], in1[k], s1);
        }
        float* q = &Aw[(size_t)idx[i] * NN + j];
        q[0] = s0;
        q[1] = s1;
      }
    }
    __syncthreads();  // phase2 reads the row-updated Aw

    // ---- phase 2: rewrite 16 cols of Aw, this thread owns rows 2t, 2t+1 --
#pragma unroll
    for (int rr = 0; rr < 2; ++rr) {
      const int r = tid * 2 + rr;
      float in[16];
#pragma unroll
      for (int k = 0; k < 16; ++k) in[k] = Aw[(size_t)r * NN + idx[k]];
#pragma unroll
      for (int i = 0; i < 16; ++i) {
        float s = 0.0f;
#pragma unroll
        for (int k = 0; k < 16; ++k) s = fmaf(in[k], Osh[i][k], s);
        Aw[(size_t)r * NN + idx[i]] = s;
      }
    }

    // ---- phase 3: rewrite 16 rows of right (disjoint from phase2 data) ---
    {
      const int j = tid * 2;
      float in0[16], in1[16];
#pragma unroll
      for (int k = 0; k < 16; ++k) {
        const float* p = &right[(size_t)idx[k] * NN + j];
        in0[k] = p[0];
        in1[k] = p[1];
      }
#pragma unroll
      for (int i = 0; i < 16; ++i) {
        float s0 = 0.0f, s1 = 0.0f;
#pragma unroll
        for (int k = 0; k < 16; ++k) {
          s0 = fmaf(Osh[i][k], in0[k], s0);
          s1 = fmaf(Osh[i][k], in1[k], s1);
        }
        float* q = &right[(size_t)idx[i] * NN + j];
        q[0] = s0;
        q[1] = s1;
      }
    }
  }
}

// ---------------------------------------------------------------------------
// mask: D[i][j] *= (i==j) ? 1 : active[i]*active[j]   (in place)
// ---------------------------------------------------------------------------
__global__ __launch_bounds__(256) void mmf_mask_kernel(float* __restrict__ D,
                                                       const float* __restrict__ active) {
  size_t i = (size_t)blockIdx.x * blockDim.x + threadIdx.x;
  if (i >= (size_t)NN * NN) return;
  int row = (int)(i / NN);
  int col = (int)(i % NN);
  float m = (row == col) ? 1.0f : active[row] * active[col];
  D[i] *= m;
}

// ---------------------------------------------------------------------------
// f32 WMMA GEMM: C = op(A) @ B, op(A)=A^T if TRANSA.  N=2048 fixed.
// One wave (32 threads) per block computes a 64x64 macro-tile (4x4 WMMA tiles),
// K stepped by 4 with V_WMMA_F32_16X16X4_F32.
//
// A 16x4 fragment layout (ISA 7.12.2): lanes 0-15 hold K=kk,kk+1 for M=lane;
// lanes 16-31 hold K=kk+2,kk+3.  B 4x16 mirrored.  C/D: VGPR r, lanes 0-15 ->
// M=r, lanes 16-31 -> M=8+r, N=lane&15.
// ---------------------------------------------------------------------------
template <int TRANSA>
__global__ __launch_bounds__(32) void mmf_gemm_kernel(const float* __restrict__ A,
                                                      const float* __restrict__ B,
                                                      float* __restrict__ C) {
  const int lane = threadIdx.x;
  const int half = lane >> 4;   // 0: K pair (0,1); 1: K pair (2,3)
  const int l15 = lane & 15;
  const int m0 = blockIdx.y * 64;
  const int n0 = blockIdx.x * 64;

  v8f acc[4][4];
  const v8f vzero = {};
#pragma unroll
  for (int mi = 0; mi < 4; ++mi)
#pragma unroll
    for (int ni = 0; ni < 4; ++ni) acc[mi][ni] = vzero;

  for (int kk = 0; kk < NN; kk += 4) {
    const int kb = kk + half * 2;  // this lane's first K
    v2f a[4], b[4];
#pragma unroll
    for (int mi = 0; mi < 4; ++mi) {
      const int m = m0 + 16 * mi + l15;
      if (TRANSA) {
        // opA[m][k] = A[k][m]  -> lane-coalesced reads of row-major A
        a[mi].x = A[(size_t)kb * NN + m];
        a[mi].y = A[(size_t)(kb + 1) * NN + m];
      } else {
        const float* p = &A[(size_t)m * NN + kb];
        a[mi].x = p[0];
        a[mi].y = p[1];
      }
    }
#pragma unroll
    for (int ni = 0; ni < 4; ++ni) {
      const int n = n0 + 16 * ni + l15;
      b[ni].x = B[(size_t)kb * NN + n];
      b[ni].y = B[(size_t)(kb + 1) * NN + n];
    }
#pragma unroll
    for (int mi = 0; mi < 4; ++mi)
#pragma unroll
      for (int ni = 0; ni < 4; ++ni)
        acc[mi][ni] = __builtin_amdgcn_wmma_f32_16x16x4_f32(
            false, a[mi], false, b[ni], (short)0, acc[mi][ni], false, false);
  }

#pragma unroll
  for (int mi = 0; mi < 4; ++mi)
#pragma unroll
    for (int ni = 0; ni < 4; ++ni) {
      const int col = n0 + 16 * ni + l15;
#pragma unroll
      for (int r = 0; r < 8; ++r) {
        const int row = m0 + 16 * mi + r + half * 8;
        C[(size_t)row * NN + col] = acc[mi][ni][r];
      }
    }
}

// ---------------------------------------------------------------------------
// Host side
// ---------------------------------------------------------------------------
extern "C" void kernel_launch(void* const* d_in, const int* in_sizes, int n_in,
                              void* d_out, int out_size, void* d_ws, size_t ws_size,
                              hipStream_t stream) {
  (void)n_in; (void)out_size; (void)ws_size;

  const float* A     = (const float*)d_in[0];
  const float* O_all = (const float*)d_in[1];
  const int*   sel   = (const int*)d_in[2];
  const int*   wav   = (const int*)d_in[3];

  const int L = in_sizes[2] / 16;       // 256 levels
  const int Lw = in_sizes[3];           // wav count

  float* out   = (float*)d_out;
  float* A_rec = out;                             // output 0
  float* right = out + (size_t)NN * NN;           // output 1
  float* Dw    = out + 2 * (size_t)NN * NN;       // output 2 (working A -> D)

  float* T1     = (float*)d_ws;                   // 16 MB scratch: right^T @ D
  float* active = T1 + (size_t)NN * NN;           // 2048 floats

  const size_t nelem = (size_t)NN * NN;
  const int eb = 256;
  const int eg = (int)((nelem + eb - 1) / eb);

  mmf_init_kernel<<<eg, eb, 0, stream>>>(A, Dw, right, active);
  mmf_wav_kernel<<<(Lw + 255) / 256, 256, 0, stream>>>(wav, active, Lw);
  mmf_levels_kernel<<<1, 1024, 0, stream>>>(Dw, right, O_all, sel, L);
  mmf_mask_kernel<<<eg, eb, 0, stream>>>(Dw, active);

  dim3 ggrid(NN / 64, NN / 64);
  mmf_gemm_kernel<1><<<ggrid, 32, 0, stream>>>(right, Dw, T1);   // T1 = right^T @ D
  mmf_gemm_kernel<0><<<ggrid, 32, 0, stream>>>(T1, right, A_rec); // A_rec = T1 @ right
}